// CGATLayer_16707422781530
// MI455X (gfx1250) — compile-verified
//
#include <hip/hip_runtime.h>

// GAT layer: N=50000 nodes, DEG=32, IN_DIM=128, OUT_DIM=64, leaky slope 0.01
#define GAT_N        50000
#define GAT_DEG      32
#define GAT_IN_DIM   128
#define GAT_OUT_DIM  64
#define GAT_SLOPE    0.01f

typedef __attribute__((ext_vector_type(2))) float v2f;
typedef __attribute__((ext_vector_type(8))) float v8f;

// ---------------------------------------------------------------------------
// Kernel 1: z = feature @ W  via V_WMMA_F32_16X16X4_F32 (wave32).
// One wave computes a 16-row block of z across all 64 output columns
// (4 C-tiles of 16x16, 8 accum VGPRs each). K=128 in steps of 4.
// A layout (16x4 f32): lanes 0-15 -> rows 0-15, K pair {0,1};
//                      lanes 16-31 -> rows 0-15, K pair {2,3}.
// B layout (4x16 f32): lanes 0-15 -> cols, K pair {0,1}; lanes 16-31 -> {2,3}.
// C layout: VGPR r holds row r (lanes 0-15) and row r+8 (lanes 16-31).
// ---------------------------------------------------------------------------
__global__ __launch_bounds__(256) void gat_gemm_z(
    const float* __restrict__ feature,   // [N, 128]
    const float* __restrict__ W,         // [128, 64]
    float* __restrict__ z,               // [N, 64]  (workspace)
    int nRowBlocks)                      // N/16
{
    const int lane  = threadIdx.x & 31;
    const int wave  = threadIdx.x >> 5;
    const int rowBlock = blockIdx.x * 8 + wave;
    if (rowBlock >= nRowBlocks) return;   // wave-uniform exit: EXEC all-1s inside

    const int half = lane >> 4;           // 0: K+{0,1}   1: K+{2,3}
    const int l16  = lane & 15;
    const int koff = half * 2;
    const int rowA = rowBlock * 16 + l16;

    const float* __restrict__ Arow = feature + (size_t)rowA * GAT_IN_DIM + koff;

    v8f acc0 = {}, acc1 = {}, acc2 = {}, acc3 = {};

    for (int k0 = 0; k0 < GAT_IN_DIM; k0 += 4) {
        // A fragment: 2 consecutive K values for this lane's row
        v2f a = *(const v2f*)(Arow + k0);
        // B fragments: W rows (k0+koff), (k0+koff+1), column tile t
        const float* __restrict__ Wr0 = W + (size_t)(k0 + koff) * GAT_OUT_DIM + l16;
        const float* __restrict__ Wr1 = Wr0 + GAT_OUT_DIM;
        v2f b0; b0.x = Wr0[ 0]; b0.y = Wr1[ 0];
        v2f b1; b1.x = Wr0[16]; b1.y = Wr1[16];
        v2f b2; b2.x = Wr0[32]; b2.y = Wr1[32];
        v2f b3; b3.x = Wr0[48]; b3.y = Wr1[48];
        acc0 = __builtin_amdgcn_wmma_f32_16x16x4_f32(false, a, false, b0, (short)0, acc0, false, false);
        acc1 = __builtin_amdgcn_wmma_f32_16x16x4_f32(false, a, false, b1, (short)0, acc1, false, false);
        acc2 = __builtin_amdgcn_wmma_f32_16x16x4_f32(false, a, false, b2, (short)0, acc2, false, false);
        acc3 = __builtin_amdgcn_wmma_f32_16x16x4_f32(false, a, false, b3, (short)0, acc3, false, false);
    }

    // Store C tiles: VGPR r -> row r (lanes 0-15) / row r+8 (lanes 16-31)
    const int rbase = rowBlock * 16 + half * 8;
    float* __restrict__ zb = z + (size_t)rbase * GAT_OUT_DIM + l16;
#pragma unroll
    for (int r = 0; r < 8; ++r) {
        float* __restrict__ zr = zb + (size_t)r * GAT_OUT_DIM;
        zr[ 0] = acc0[r];
        zr[16] = acc1[r];
        zr[32] = acc2[r];
        zr[48] = acc3[r];
    }
}

// ---------------------------------------------------------------------------
// Kernel 2: s[n] = z[n] . attn[0:64]  ;  d[n] = z[n] . attn[64:128]
// One wave per node; lane l holds z[n, 2l:2l+2]; wave32 xor-shuffle reduction.
// ---------------------------------------------------------------------------
__global__ __launch_bounds__(256) void gat_attn_proj(
    const float* __restrict__ z,        // [N, 64]
    const float* __restrict__ attn,     // [128]
    float* __restrict__ s,              // [N]
    float* __restrict__ d,              // [N]
    int nNodes)
{
    const int node = blockIdx.x * 8 + (threadIdx.x >> 5);
    if (node >= nNodes) return;
    const int lane = threadIdx.x & 31;

    v2f zz = *(const v2f*)(z + (size_t)node * GAT_OUT_DIM + lane * 2);
    float sp = zz.x * attn[lane * 2]      + zz.y * attn[lane * 2 + 1];
    float dp = zz.x * attn[64 + lane * 2] + zz.y * attn[64 + lane * 2 + 1];
#pragma unroll
    for (int off = 16; off > 0; off >>= 1) {
        sp += __shfl_xor(sp, off, 32);
        dp += __shfl_xor(dp, off, 32);
    }
    if (lane == 0) { s[node] = sp; d[node] = dp; }
}

// ---------------------------------------------------------------------------
// Kernel 3: per destination node n (one wave each):
//   e_j     = leaky_relu(s[idx_j] + d[n])                    (lane j = edge j)
//   alpha_j = sum_k relu(e_j - e_k)                          (32 shuffles)
//   h[n]    = sum_j alpha_j * z[idx_j]   (lane owns channels 2l, 2l+1 -> one
//                                         global_load_b64 per neighbor row)
// The 32 row addresses are known at wave entry, so each lane prefetches both
// 128B lines of its own neighbor's row (global_prefetch_b8, no counter cost)
// before the shuffle-serialized gather loop. z is L2-resident (12.8 MB <<
// 192 MB L2), so the ~410 MB of gather traffic never touches HBM.
// ---------------------------------------------------------------------------
__global__ __launch_bounds__(256) void gat_aggregate(
    const int*   __restrict__ src_idx,  // [N, 32]
    const float* __restrict__ z,        // [N, 64]
    const float* __restrict__ s,        // [N]
    const float* __restrict__ d,        // [N]
    float* __restrict__ h,              // [N, 64]  (output)
    int nNodes)
{
    const int node = blockIdx.x * 8 + (threadIdx.x >> 5);
    if (node >= nNodes) return;
    const int lane = threadIdx.x & 31;

    const int idx = src_idx[(size_t)node * GAT_DEG + lane];

    // Warm caches for the gather loop: lane j covers row idx_j (2 x 128B lines)
    const float* __restrict__ myRow = z + (size_t)idx * GAT_OUT_DIM;
    __builtin_prefetch(myRow, 0, 3);
    __builtin_prefetch(myRow + 32, 0, 3);

    float e = s[idx] + d[node];
    e = (e > 0.0f) ? e : e * GAT_SLOPE;            // leaky_relu

    float alpha = 0.0f;
#pragma unroll
    for (int k = 0; k < GAT_DEG; ++k) {
        float ek = __shfl(e, k, 32);
        float df = e - ek;
        alpha += (df > 0.0f) ? df : 0.0f;          // relu(e_j - e_k)
    }

    float hx = 0.0f, hy = 0.0f;
#pragma unroll 8
    for (int j = 0; j < GAT_DEG; ++j) {
        float aj = __shfl(alpha, j, 32);
        int   ij = __shfl(idx,   j, 32);
        v2f zv = *(const v2f*)(z + (size_t)ij * GAT_OUT_DIM + 2 * lane);
        hx += aj * zv.x;
        hy += aj * zv.y;
    }
    v2f out; out.x = hx; out.y = hy;
    *(v2f*)(h + (size_t)node * GAT_OUT_DIM + 2 * lane) = out;
}

// ---------------------------------------------------------------------------
extern "C" void kernel_launch(void* const* d_in, const int* in_sizes, int n_in,
                              void* d_out, int out_size, void* d_ws, size_t ws_size,
                              hipStream_t stream) {
    const float* feature = (const float*)d_in[0];   // [N,128] f32
    const int*   src_idx = (const int*)  d_in[1];   // [N,32]  i32
    const float* fc_w    = (const float*)d_in[2];   // [128,64] f32
    const float* attn_w  = (const float*)d_in[3];   // [128,1] f32
    float*       h       = (float*)d_out;           // [N,64] f32

    const int N = in_sizes[0] / GAT_IN_DIM;         // 50000

    // Workspace layout: z [N*64] | s [N] | d [N]
    float* z = (float*)d_ws;
    float* s = z + (size_t)N * GAT_OUT_DIM;
    float* d = s + N;

    const int nRowBlocks = (N + 15) / 16;           // 3125
    const int gemmBlocks = (nRowBlocks + 7) / 8;    // 8 waves (16-row blocks) per 256-thr block
    const int nodeBlocks = (N + 7) / 8;             // 8 nodes per 256-thr block

    gat_gemm_z   <<<gemmBlocks, 256, 0, stream>>>(feature, fc_w, z, nRowBlocks);
    gat_attn_proj<<<nodeBlocks, 256, 0, stream>>>(z, attn_w, s, d, N);
    gat_aggregate<<<nodeBlocks, 256, 0, stream>>>(src_idx, z, s, d, h, N);
}